// ClinicalBrainLLM_41231686041788
// MI455X (gfx1250) — compile-verified
//
#include <hip/hip_runtime.h>
#include <hip/hip_bf16.h>
#include <math.h>

#define B_   16
#define T_   100
#define R_   200
#define D_   128
#define H_   4
#define DH_  32
#define FF_  2048
#define HID_ 4096
#define V_   32000
#define S_   512
#define NQ_  8
#define NL_  2
#define POS_ (NQ_ + S_)   // 520

typedef __attribute__((ext_vector_type(2))) float v2f;
typedef __attribute__((ext_vector_type(8))) float v8f;

// ---------------------------------------------------------------------------
// helpers
// ---------------------------------------------------------------------------
__device__ inline float fixnum(float x) {
    if (x != x) return 0.0f;                       // NaN -> 0
    const float FMAX = 3.4028234663852886e38f;
    if (x >  FMAX) return  FMAX;                   // +inf -> max
    if (x < -FMAX) return -FMAX;                   // -inf -> lowest
    return x;
}

__device__ inline float block_reduce_sum(float v, float* red) {
    red[threadIdx.x] = v; __syncthreads();
    for (int s = blockDim.x >> 1; s > 0; s >>= 1) {
        if ((int)threadIdx.x < s) red[threadIdx.x] += red[threadIdx.x + s];
        __syncthreads();
    }
    float r = red[0]; __syncthreads();
    return r;
}

__device__ inline float block_reduce_max(float v, float* red) {
    red[threadIdx.x] = v; __syncthreads();
    for (int s = blockDim.x >> 1; s > 0; s >>= 1) {
        if ((int)threadIdx.x < s) red[threadIdx.x] = fmaxf(red[threadIdx.x], red[threadIdx.x + s]);
        __syncthreads();
    }
    float r = red[0]; __syncthreads();
    return r;
}

#define WMMA_F32(a, b, c) \
    __builtin_amdgcn_wmma_f32_16x16x4_f32(false, (a), false, (b), (short)0, (c), false, false)

// ---------------------------------------------------------------------------
// WMMA fp32 GEMM:  C[M,N] = act( A[M,K] @ W[N,K]^T + bias[N] ) (+ resid)
// one wave per block; wave computes a 32x64 output tile: 2 A fragments x
// 4 B fragments -> 8 WMMAs per K-step on 6 loads. Inner loop is software-
// pipelined: k+4 fragments are fetched before the k WMMAs issue, so the
// scheduler can overlap global-load latency with the matrix pipe instead of
// the load -> s_wait_loadcnt 0 -> wmma serialization seen previously.
// Requires: M%32==0, N%64==0, K%4==0 (true for all call sites here).
// ---------------------------------------------------------------------------
__global__ __launch_bounds__(32)
void gemm_wmma_f32(const float* __restrict__ A, const float* __restrict__ W,
                   const float* __restrict__ bias, const float* __restrict__ resid,
                   float* __restrict__ C, int M, int N, int K, int relu)
{
    const int lane = threadIdx.x & 31;
    const int half = lane >> 4;        // 0: K=0,1 / M=v   1: K=2,3 / M=v+8
    const int l16  = lane & 15;
    const int n0   = blockIdx.x * 64;
    const int m0   = blockIdx.y * 32;

    v8f acc[8] = {};                   // [mt*4 + nt]

    const float* a0 = A + (size_t)(m0 +  0 + l16) * K + 2 * half;
    const float* a1 = A + (size_t)(m0 + 16 + l16) * K + 2 * half;
    const float* w0 = W + (size_t)(n0 +  0 + l16) * K + 2 * half;
    const float* w1 = W + (size_t)(n0 + 16 + l16) * K + 2 * half;
    const float* w2 = W + (size_t)(n0 + 32 + l16) * K + 2 * half;
    const float* w3 = W + (size_t)(n0 + 48 + l16) * K + 2 * half;

    v2f ca0 = *(const v2f*)(a0);
    v2f ca1 = *(const v2f*)(a1);
    v2f cb0 = *(const v2f*)(w0);
    v2f cb1 = *(const v2f*)(w1);
    v2f cb2 = *(const v2f*)(w2);
    v2f cb3 = *(const v2f*)(w3);

    int k = 0;
    for (; k + 4 < K; k += 4) {
        const int kn = k + 4;
        // prefetch next K-step fragments (independent of current WMMAs)
        v2f na0 = *(const v2f*)(a0 + kn);
        v2f na1 = *(const v2f*)(a1 + kn);
        v2f nb0 = *(const v2f*)(w0 + kn);
        v2f nb1 = *(const v2f*)(w1 + kn);
        v2f nb2 = *(const v2f*)(w2 + kn);
        v2f nb3 = *(const v2f*)(w3 + kn);

        acc[0] = WMMA_F32(ca0, cb0, acc[0]);
        acc[1] = WMMA_F32(ca0, cb1, acc[1]);
        acc[2] = WMMA_F32(ca0, cb2, acc[2]);
        acc[3] = WMMA_F32(ca0, cb3, acc[3]);
        acc[4] = WMMA_F32(ca1, cb0, acc[4]);
        acc[5] = WMMA_F32(ca1, cb1, acc[5]);
        acc[6] = WMMA_F32(ca1, cb2, acc[6]);
        acc[7] = WMMA_F32(ca1, cb3, acc[7]);

        ca0 = na0; ca1 = na1;
        cb0 = nb0; cb1 = nb1; cb2 = nb2; cb3 = nb3;
    }
    // final (peeled) K-step
    acc[0] = WMMA_F32(ca0, cb0, acc[0]);
    acc[1] = WMMA_F32(ca0, cb1, acc[1]);
    acc[2] = WMMA_F32(ca0, cb2, acc[2]);
    acc[3] = WMMA_F32(ca0, cb3, acc[3]);
    acc[4] = WMMA_F32(ca1, cb0, acc[4]);
    acc[5] = WMMA_F32(ca1, cb1, acc[5]);
    acc[6] = WMMA_F32(ca1, cb2, acc[6]);
    acc[7] = WMMA_F32(ca1, cb3, acc[7]);

#pragma unroll
    for (int mt = 0; mt < 2; ++mt) {
#pragma unroll
        for (int nt = 0; nt < 4; ++nt) {
            const int ncol = n0 + nt * 16 + l16;
            const float bv = bias ? bias[ncol] : 0.0f;
#pragma unroll
            for (int v = 0; v < 8; ++v) {
                const int row = m0 + mt * 16 + v + 8 * half;
                float val = acc[mt * 4 + nt][v] + bv;
                if (relu) val = fmaxf(val, 0.0f);
                const size_t idx = (size_t)row * N + ncol;
                if (resid) val += resid[idx];
                C[idx] = val;
            }
        }
    }
}

// ---------------------------------------------------------------------------
// conv1: depthwise (groups=R), in [B,R,T] (from bold [B,T,R], nan_to_num),
// out [B,2R,T]; weight [2R,1,3], pad 1
// ---------------------------------------------------------------------------
__global__ __launch_bounds__(256)
void conv1_kernel(const float* __restrict__ bold, const float* __restrict__ w,
                  const float* __restrict__ bias, float* __restrict__ out)
{
    int idx = blockIdx.x * blockDim.x + threadIdx.x;
    if (idx >= B_ * 2 * R_ * T_) return;
    int t = idx % T_;
    int c = (idx / T_) % (2 * R_);
    int b = idx / (T_ * 2 * R_);
    int r = c >> 1;                    // input channel of this group
    float acc = bias[c];
#pragma unroll
    for (int j = 0; j < 3; ++j) {
        int tt = t + j - 1;
        if (tt >= 0 && tt < T_) {
            float xv = fixnum(bold[((size_t)b * T_ + tt) * R_ + r]);
            acc += w[c * 3 + j] * xv;
        }
    }
    out[idx] = acc;
}

// conv2: groups=R, in [B,2R,T], out [B,R,T]; weight [R,2,3]
__global__ __launch_bounds__(256)
void conv2_kernel(const float* __restrict__ x, const float* __restrict__ w,
                  const float* __restrict__ bias, float* __restrict__ out)
{
    int idx = blockIdx.x * blockDim.x + threadIdx.x;
    if (idx >= B_ * R_ * T_) return;
    int t = idx % T_;
    int r = (idx / T_) % R_;
    int b = idx / (T_ * R_);
    float acc = bias[r];
#pragma unroll
    for (int i = 0; i < 2; ++i) {
        const float* xc = x + ((size_t)b * (2 * R_) + 2 * r + i) * T_;
#pragma unroll
        for (int j = 0; j < 3; ++j) {
            int tt = t + j - 1;
            if (tt >= 0 && tt < T_) acc += w[(r * 2 + i) * 3 + j] * xc[tt];
        }
    }
    out[idx] = acc;
}

// per-channel batch stats over (B,T) for x [B,C,T] -> stats[c] = {mean, rstd}
__global__ __launch_bounds__(256)
void bnstats_kernel(const float* __restrict__ x, float* __restrict__ stats, int C)
{
    __shared__ float red[256];
    int c = blockIdx.x;
    float s = 0.f, ss = 0.f;
    for (int i = threadIdx.x; i < B_ * T_; i += 256) {
        int b = i / T_, t = i % T_;
        float v = x[((size_t)b * C + c) * T_ + t];
        s += v; ss += v * v;
    }
    float tot  = block_reduce_sum(s,  red);
    float tot2 = block_reduce_sum(ss, red);
    if (threadIdx.x == 0) {
        const float n = (float)(B_ * T_);
        float m   = tot / n;
        float var = tot2 / n - m * m;
        stats[2 * c]     = m;
        stats[2 * c + 1] = rsqrtf(var + 1e-5f);
    }
}

__global__ __launch_bounds__(256)
void bnrelu_kernel(float* __restrict__ x, const float* __restrict__ stats,
                   const float* __restrict__ g, const float* __restrict__ bta, int C)
{
    int idx = blockIdx.x * blockDim.x + threadIdx.x;
    if (idx >= B_ * C * T_) return;
    int c = (idx / T_) % C;
    float v = (x[idx] - stats[2 * c]) * stats[2 * c + 1] * g[c] + bta[c];
    x[idx] = fmaxf(v, 0.0f);
}

// ---------------------------------------------------------------------------
// graph attention: z[b,q,:] = softmax(Q[b,q]·K[b,k] / sqrt(D)) @ h[b]
// one block per (b,q)
// ---------------------------------------------------------------------------
__global__ __launch_bounds__(256)
void graph_attn_kernel(const float* __restrict__ Q, const float* __restrict__ Km,
                       const float* __restrict__ h, float* __restrict__ z)
{
    __shared__ float sc[R_];
    __shared__ float red[256];
    int b = blockIdx.x / R_, q = blockIdx.x % R_;
    const float scale = rsqrtf((float)D_);
    const float* qrow = Q + ((size_t)b * R_ + q) * D_;
    for (int k = threadIdx.x; k < R_; k += 256) {
        const float* krow = Km + ((size_t)b * R_ + k) * D_;
        float s = 0.f;
        for (int d = 0; d < D_; ++d) s += qrow[d] * krow[d];
        sc[k] = s * scale;
    }
    __syncthreads();
    float mx = -3.4e38f;
    for (int k = threadIdx.x; k < R_; k += 256) mx = fmaxf(mx, sc[k]);
    mx = block_reduce_max(mx, red);
    float sum = 0.f;
    for (int k = threadIdx.x; k < R_; k += 256) { float e = __expf(sc[k] - mx); sc[k] = e; sum += e; }
    sum = block_reduce_sum(sum, red);
    float inv = 1.0f / sum;
    if (threadIdx.x < D_) {
        int d = threadIdx.x;
        float acc = 0.f;
        for (int k = 0; k < R_; ++k) acc += sc[k] * h[((size_t)b * R_ + k) * D_ + d];
        z[((size_t)b * R_ + q) * D_ + d] = acc * inv;
    }
}

// multi-head attention over qkv [B,R,3D]; one block per (b,h,q); o [B,R,D]
__global__ __launch_bounds__(256)
void mha_kernel(const float* __restrict__ qkv, float* __restrict__ o)
{
    __shared__ float sc[R_];
    __shared__ float red[256];
    int id = blockIdx.x;
    int q  = id % R_;
    int hh = (id / R_) % H_;
    int b  = id / (R_ * H_);
    const float scale = rsqrtf((float)DH_);
    const float* qrow = qkv + ((size_t)b * R_ + q) * (3 * D_) + hh * DH_;
    for (int k = threadIdx.x; k < R_; k += 256) {
        const float* krow = qkv + ((size_t)b * R_ + k) * (3 * D_) + D_ + hh * DH_;
        float s = 0.f;
#pragma unroll
        for (int d = 0; d < DH_; ++d) s += qrow[d] * krow[d];
        sc[k] = s * scale;
    }
    __syncthreads();
    float mx = -3.4e38f;
    for (int k = threadIdx.x; k < R_; k += 256) mx = fmaxf(mx, sc[k]);
    mx = block_reduce_max(mx, red);
    float sum = 0.f;
    for (int k = threadIdx.x; k < R_; k += 256) { float e = __expf(sc[k] - mx); sc[k] = e; sum += e; }
    sum = block_reduce_sum(sum, red);
    float inv = 1.0f / sum;
    if (threadIdx.x < DH_) {
        int d = threadIdx.x;
        float acc = 0.f;
        for (int k = 0; k < R_; ++k)
            acc += sc[k] * qkv[((size_t)b * R_ + k) * (3 * D_) + 2 * D_ + hh * DH_ + d];
        o[((size_t)b * R_ + q) * D_ + hh * DH_ + d] = acc * inv;
    }
}

// pooled attention: ctx[b,q,:] = softmax(qt[q]·z[b,k]/sqrt(D)) @ z[b]
__global__ __launch_bounds__(256)
void pool_attn_kernel(const float* __restrict__ qt, const float* __restrict__ z,
                      float* __restrict__ ctx)
{
    __shared__ float sc[R_];
    __shared__ float red[256];
    int q = blockIdx.x % NQ_, b = blockIdx.x / NQ_;
    const float scale = rsqrtf((float)D_);
    const float* qrow = qt + (size_t)q * D_;
    for (int k = threadIdx.x; k < R_; k += 256) {
        const float* krow = z + ((size_t)b * R_ + k) * D_;
        float s = 0.f;
        for (int d = 0; d < D_; ++d) s += qrow[d] * krow[d];
        sc[k] = s * scale;
    }
    __syncthreads();
    float mx = -3.4e38f;
    for (int k = threadIdx.x; k < R_; k += 256) mx = fmaxf(mx, sc[k]);
    mx = block_reduce_max(mx, red);
    float sum = 0.f;
    for (int k = threadIdx.x; k < R_; k += 256) { float e = __expf(sc[k] - mx); sc[k] = e; sum += e; }
    sum = block_reduce_sum(sum, red);
    float inv = 1.0f / sum;
    if (threadIdx.x < D_) {
        int d = threadIdx.x;
        float acc = 0.f;
        for (int k = 0; k < R_; ++k) acc += sc[k] * z[((size_t)b * R_ + k) * D_ + d];
        ctx[((size_t)b * NQ_ + q) * D_ + d] = acc * inv;
    }
}

// row-wise layernorm: one block per row of width Ncols
__global__ __launch_bounds__(256)
void layernorm_kernel(const float* __restrict__ x, const float* __restrict__ g,
                      const float* __restrict__ bta, float* __restrict__ y, int Ncols)
{
    __shared__ float red[256];
    const float* xr = x + (size_t)blockIdx.x * Ncols;
    float* yr       = y + (size_t)blockIdx.x * Ncols;
    float s = 0.f;
    for (int i = threadIdx.x; i < Ncols; i += 256) s += xr[i];
    float m = block_reduce_sum(s, red) / (float)Ncols;
    float v = 0.f;
    for (int i = threadIdx.x; i < Ncols; i += 256) { float d = xr[i] - m; v += d * d; }
    float inv = rsqrtf(block_reduce_sum(v, red) / (float)Ncols + 1e-5f);
    for (int i = threadIdx.x; i < Ncols; i += 256)
        yr[i] = (xr[i] - m) * inv * g[i] + bta[i];
}

// assemble inputs_embeds: row = b*520+pos; brain rows for pos<NQ, else table
// gather. This is the HBM-bound stage (~270 MB total traffic); use b128
// vector copies and prefetch the next row's source while copying this one.
__global__ __launch_bounds__(256)
void assemble_kernel(const float* __restrict__ brain, const int* __restrict__ ids,
                     const float* __restrict__ table, float* __restrict__ out)
{
    int row = blockIdx.x;
    int pos = row % POS_, b = row / POS_;
    const float* src = (pos < NQ_)
        ? brain + ((size_t)b * NQ_ + pos) * HID_
        : table + (size_t)ids[b * S_ + (pos - NQ_)] * HID_;

    // prefetch the source row of the *next* position (gfx1250 global_prefetch_b8)
    int nrow = row + 1;
    if (nrow < B_ * POS_) {
        int npos = nrow % POS_, nb = nrow / POS_;
        const float* nsrc = (npos < NQ_)
            ? brain + ((size_t)nb * NQ_ + npos) * HID_
            : table + (size_t)ids[nb * S_ + (npos - NQ_)] * HID_;
        __builtin_prefetch(nsrc + threadIdx.x * (HID_ / 256), 0, 1);
    }

    float4* dst = (float4*)(out + (size_t)row * HID_);
    const float4* s4 = (const float4*)src;
    for (int i = threadIdx.x; i < HID_ / 4; i += 256) dst[i] = s4[i];
}

__global__ __launch_bounds__(256)
void masklab_kernel(const int* __restrict__ amask, const int* __restrict__ lab,
                    float* __restrict__ out_mask, float* __restrict__ out_lab)
{
    int i = blockIdx.x * blockDim.x + threadIdx.x;
    if (i >= B_ * POS_) return;
    int pos = i % POS_, b = i / POS_;
    float mv, lv;
    if (pos < NQ_) { mv = 1.0f; lv = -100.0f; }
    else {
        mv = (float)amask[b * S_ + pos - NQ_];
        lv = (float)lab[b * S_ + pos - NQ_];
    }
    out_mask[i] = mv;
    out_lab[i]  = lv;
}

// ---------------------------------------------------------------------------
extern "C" void kernel_launch(void* const* d_in, const int* in_sizes, int n_in,
                              void* d_out, int out_size, void* d_ws, size_t ws_size,
                              hipStream_t stream)
{
    const float* bold      = (const float*)d_in[0];
    const int*   input_ids = (const int*)  d_in[1];
    const int*   attn_mask = (const int*)  d_in[2];
    const int*   labels    = (const int*)  d_in[3];
    const float* conv1_w = (const float*)d_in[4];
    const float* conv1_b = (const float*)d_in[5];
    const float* bn1_g   = (const float*)d_in[6];
    const float* bn1_b   = (const float*)d_in[7];
    const float* conv2_w = (const float*)d_in[8];
    const float* conv2_b = (const float*)d_in[9];
    const float* bn2_g   = (const float*)d_in[10];
    const float* bn2_b   = (const float*)d_in[11];
    const float* tproj_w = (const float*)d_in[12];
    const float* tproj_b = (const float*)d_in[13];
    const float* wq_w    = (const float*)d_in[14];
    const float* wq_b    = (const float*)d_in[15];
    const float* wk_w    = (const float*)d_in[16];
    const float* wk_b    = (const float*)d_in[17];
    const float* ln1_g   = (const float*)d_in[18];
    const float* ln1_b   = (const float*)d_in[19];
    const float* qkv_w   = (const float*)d_in[20];
    const float* qkv_b   = (const float*)d_in[21];
    const float* out_w   = (const float*)d_in[22];
    const float* out_b   = (const float*)d_in[23];
    const float* ln2_g   = (const float*)d_in[24];
    const float* ln2_b   = (const float*)d_in[25];
    const float* ff1_w   = (const float*)d_in[26];
    const float* ff1_b   = (const float*)d_in[27];
    const float* ff2_w   = (const float*)d_in[28];
    const float* ff2_b   = (const float*)d_in[29];
    const float* proj_w  = (const float*)d_in[30];
    const float* proj_b  = (const float*)d_in[31];
    const float* lnf_g   = (const float*)d_in[32];
    const float* lnf_b   = (const float*)d_in[33];
    const float* qtok    = (const float*)d_in[34];
    const float* table   = (const float*)d_in[35];

    float* ws = (float*)d_ws;
    float* x1     = ws;                    // [B,2R,T]   640000
    float* stats1 = x1     + 640000;       // [2R,2]        800
    float* x2     = stats1 + 800;          // [B,R,T]    320000
    float* stats2 = x2     + 320000;       // [R,2]         400
    float* h      = stats2 + 400;          // [B,R,D]    409600
    float* Qm     = h      + 409600;       // [B,R,D]
    float* Km     = Qm     + 409600;       // [B,R,D]
    float* z      = Km     + 409600;       // [B,R,D]
    float* y      = z      + 409600;       // [B,R,D]
    float* qkv    = y      + 409600;       // [B,R,3D]  1228800
    float* o      = qkv    + 1228800;      // [B,R,D]
    float* ff     = o      + 409600;       // [B,R,FF]  6553600
    float* ctx    = ff     + 6553600;      // [B,NQ,D]    16384
    float* pe     = ctx    + 16384;        // [B,NQ,HID] 524288
    float* brain  = pe     + 524288;       // [B,NQ,HID]

    const int M = B_ * R_;                 // 3200 rows for the big GEMMs
    dim3 blk256(256), blk32(32);

    // --- TemporalGraphEncoder ---
    conv1_kernel<<<(B_*2*R_*T_ + 255)/256, blk256, 0, stream>>>(bold, conv1_w, conv1_b, x1);
    bnstats_kernel<<<2*R_, blk256, 0, stream>>>(x1, stats1, 2*R_);
    bnrelu_kernel<<<(B_*2*R_*T_ + 255)/256, blk256, 0, stream>>>(x1, stats1, bn1_g, bn1_b, 2*R_);
    conv2_kernel<<<(B_*R_*T_ + 255)/256, blk256, 0, stream>>>(x1, conv2_w, conv2_b, x2);
    bnstats_kernel<<<R_, blk256, 0, stream>>>(x2, stats2, R_);
    bnrelu_kernel<<<(B_*R_*T_ + 255)/256, blk256, 0, stream>>>(x2, stats2, bn2_g, bn2_b, R_);

    // h = x2 @ tproj_w^T + tproj_b    [3200,100] x [128,100]^T
    gemm_wmma_f32<<<dim3(D_/64, M/32), blk32, 0, stream>>>(x2, tproj_w, tproj_b, nullptr, h, M, D_, T_, 0);
    // Q,K projections
    gemm_wmma_f32<<<dim3(D_/64, M/32), blk32, 0, stream>>>(h, wq_w, wq_b, nullptr, Qm, M, D_, D_, 0);
    gemm_wmma_f32<<<dim3(D_/64, M/32), blk32, 0, stream>>>(h, wk_w, wk_b, nullptr, Km, M, D_, D_, 0);
    // adjacency softmax + z = adj @ h
    graph_attn_kernel<<<B_*R_, blk256, 0, stream>>>(Qm, Km, h, z);

    // --- 2-layer pre-norm transformer ---
    for (int l = 0; l < NL_; ++l) {
        layernorm_kernel<<<M, blk256, 0, stream>>>(z, ln1_g + l*D_, ln1_b + l*D_, y, D_);
        gemm_wmma_f32<<<dim3(3*D_/64, M/32), blk32, 0, stream>>>(
            y, qkv_w + (size_t)l*3*D_*D_, qkv_b + (size_t)l*3*D_, nullptr, qkv, M, 3*D_, D_, 0);
        mha_kernel<<<B_*H_*R_, blk256, 0, stream>>>(qkv, o);
        gemm_wmma_f32<<<dim3(D_/64, M/32), blk32, 0, stream>>>(
            o, out_w + (size_t)l*D_*D_, out_b + (size_t)l*D_, z, z, M, D_, D_, 0);
        layernorm_kernel<<<M, blk256, 0, stream>>>(z, ln2_g + l*D_, ln2_b + l*D_, y, D_);
        gemm_wmma_f32<<<dim3(FF_/64, M/32), blk32, 0, stream>>>(
            y, ff1_w + (size_t)l*FF_*D_, ff1_b + (size_t)l*FF_, nullptr, ff, M, FF_, D_, 1);
        gemm_wmma_f32<<<dim3(D_/64, M/32), blk32, 0, stream>>>(
            ff, ff2_w + (size_t)l*D_*FF_, ff2_b + (size_t)l*D_, z, z, M, D_, FF_, 0);
    }

    // --- SDPA pooling + projector + final LN ---
    pool_attn_kernel<<<B_*NQ_, blk256, 0, stream>>>(qtok, z, ctx);
    gemm_wmma_f32<<<dim3(HID_/64, (B_*NQ_)/32), blk32, 0, stream>>>(
        ctx, proj_w, proj_b, nullptr, pe, B_*NQ_, HID_, D_, 0);
    layernorm_kernel<<<B_*NQ_, blk256, 0, stream>>>(pe, lnf_g, lnf_b, brain, HID_);

    // --- embedding gather + concat + mask/labels ---
    float* out_embeds = (float*)d_out;
    float* out_mask   = out_embeds + (size_t)B_ * POS_ * HID_;
    float* out_lab    = out_mask + B_ * POS_;
    assemble_kernel<<<B_*POS_, blk256, 0, stream>>>(brain, input_ids, table, out_embeds);
    masklab_kernel<<<(B_*POS_ + 255)/256, blk256, 0, stream>>>(attn_mask, labels, out_mask, out_lab);
}